// high_order_input_5506148073824
// MI455X (gfx1250) — compile-verified
//
#include <hip/hip_runtime.h>
#include <stdint.h>

typedef float v4f __attribute__((ext_vector_type(4)));

#define IMG_H 128
#define IMG_W 128
#define IMG_L (IMG_H * IMG_W)          // 16384
#define ROWS_PER_BLOCK 8
#define BANDS (IMG_H / ROWS_PER_BLOCK) // 16
#define BAND_ROWS (ROWS_PER_BLOCK + 2) // 10 (halo top+bottom)
#define LDS_STRIDE 136                 // floats/row; data in cols 4..131, pads at 3 and 132
#define NTERMS 210                     // 45 order-2 + 165 order-3

__device__ __forceinline__ void async_copy_g2l(float* lds_dst, const float* g_src) {
#if defined(__gfx1250__)
    // GLOBAL_LOAD_ASYNC_TO_LDS_B32: copies one dword from global memory into LDS; tracked by ASYNCcnt.
    uint32_t lds_addr = (uint32_t)(uintptr_t)lds_dst;   // flat LDS aperture: offset in addr[31:0]
    uint64_t gaddr    = (uint64_t)(uintptr_t)g_src;
    asm volatile("global_load_async_to_lds_b32 %0, %1, off"
                 :
                 : "v"(lds_addr), "v"(gaddr)
                 : "memory");
#else
    *lds_dst = *g_src;
#endif
}

__device__ __forceinline__ void wait_async_zero() {
#if defined(__gfx1250__)
#if __has_builtin(__builtin_amdgcn_s_wait_asynccnt)
    __builtin_amdgcn_s_wait_asynccnt(0);
#else
    asm volatile("s_wait_asynccnt 0x0" ::: "memory");
#endif
#endif
}

__global__ __launch_bounds__(256)
void high_order_monomial_kernel(const float* __restrict__ x, float* __restrict__ out) {
    __shared__ float smem[BAND_ROWS * LDS_STRIDE];

    const int tid  = threadIdx.x;
    const int bc   = blockIdx.x >> 4;       // image index (B*C flattened)
    const int band = blockIdx.x & (BANDS - 1);
    const int h0   = band * ROWS_PER_BLOCK;

    const float* __restrict__ xim = x + (size_t)bc * IMG_L;

    // Zero the left/right pad columns (cols 3 and 132) of each band row.
    if (tid < BAND_ROWS * 2) {
        int lr  = tid >> 1;
        int col = (tid & 1) ? (4 + IMG_W) : 3;
        smem[lr * LDS_STRIDE + col] = 0.0f;
    }

    // Stage BAND_ROWS x 128 floats into LDS via async global->LDS DMA.
    // 1280 elements, 256 threads -> 5 uniform iterations; each wave's 32
    // contiguous indices stay inside one row, so the OOB-row branch is
    // wave-uniform.
    #pragma unroll
    for (int base = 0; base < BAND_ROWS * IMG_W; base += 256) {
        int idx = base + tid;
        int lr  = idx >> 7;            // 0..9
        int cc  = idx & (IMG_W - 1);   // 0..127
        int g   = h0 - 1 + lr;         // global row (with halo)
        float* dst = &smem[lr * LDS_STRIDE + 4 + cc];
        if (g >= 0 && g < IMG_H) {
            async_copy_g2l(dst, xim + (size_t)g * IMG_W + cc);
        } else {
            *dst = 0.0f;               // top/bottom zero padding rows
        }
    }

    wait_async_zero();
    __syncthreads();

    // Each thread computes 4 adjacent pixels (float4 lanes).
    const int r  = tid >> 5;              // 0..7 row within band
    const int w0 = (tid & 31) * 4;        // 0,4,...,124
    const int h  = h0 + r;

    // Build the 9 neighborhood vectors a[kh*3+kw]; element p corresponds to
    // pixel (h, w0+p): a[k][p] = xpad[h+kh, w0+p+kw] = smem[r+kh][4 + w0+p+kw-1].
    v4f a[9];
    #pragma unroll
    for (int dr = 0; dr < 3; ++dr) {
        const float* row = &smem[(r + dr) * LDS_STRIDE];
        v4f   m   = *(const v4f*)&row[4 + w0];   // 16B-aligned ds_load_b128
        float lft = row[3 + w0];
        float rgt = row[8 + w0];
        v4f t0; t0.x = lft; t0.y = m.x; t0.z = m.y; t0.w = m.z;   // kw=0 (w-1)
        v4f t2; t2.x = m.y; t2.y = m.z; t2.z = m.w; t2.w = rgt;   // kw=2 (w+1)
        a[dr * 3 + 0] = t0;
        a[dr * 3 + 1] = m;                                        // kw=1 (w)
        a[dr * 3 + 2] = t2;
    }

    // Output layout: [bc][term][h*W + w]; terms 0..44 = pairs (i<=j),
    // 45..209 = triples (i<=j<=k), both lexicographic — matches tpcm2/tpcm3.
    float* outp = out + (size_t)bc * NTERMS * IMG_L + (size_t)h * IMG_W + w0;

    int t2i = 0;   // order-2 slot
    int t3i = 45;  // order-3 slot
    #pragma unroll
    for (int i = 0; i < 9; ++i) {
        #pragma unroll
        for (int j = i; j < 9; ++j) {
            v4f pij = a[i] * a[j];
            __builtin_nontemporal_store(pij, (v4f*)(outp + (size_t)t2i * IMG_L));
            ++t2i;
            #pragma unroll
            for (int k = j; k < 9; ++k) {
                v4f p3 = pij * a[k];
                __builtin_nontemporal_store(p3, (v4f*)(outp + (size_t)t3i * IMG_L));
                ++t3i;
            }
        }
    }
}

extern "C" void kernel_launch(void* const* d_in, const int* in_sizes, int n_in,
                              void* d_out, int out_size, void* d_ws, size_t ws_size,
                              hipStream_t stream) {
    (void)n_in; (void)d_ws; (void)ws_size; (void)out_size;
    const float* x = (const float*)d_in[0];
    float* out = (float*)d_out;
    // tpcm2/tpcm3 (d_in[1], d_in[2]) are the canonical lexicographic index
    // tables; the kernel regenerates them at compile time, so they are unused.
    int n_images = in_sizes[0] / IMG_L;    // B*C = 32
    dim3 grid(n_images * BANDS);
    dim3 block(256);
    hipLaunchKernelGGL(high_order_monomial_kernel, grid, block, 0, stream, x, out);
}